// PooledGCNSegmentation_35588099015558
// MI455X (gfx1250) — compile-verified
//
#include <hip/hip_runtime.h>
#include <hip/hip_bf16.h>

typedef float v2f __attribute__((ext_vector_type(2)));
typedef float v8f __attribute__((ext_vector_type(8)));

#define HID   128
#define NCLS  18
#define KSEL  5000
#define SORTN 16384   // pow2 >= nodes-per-graph (12500)

// ---------------- utility ----------------
__global__ void zero_u32_kernel(unsigned* __restrict__ p, int n) {
    int i = blockIdx.x * blockDim.x + threadIdx.x;
    if (i < n) p[i] = 0u;
}

__global__ void deg_kernel(const int* __restrict__ src, const int* __restrict__ dst,
                           unsigned* __restrict__ dout, unsigned* __restrict__ din, int E) {
    int e = blockIdx.x * blockDim.x + threadIdx.x;
    if (e < E) {
        atomicAdd(&dout[src[e]], 1u);
        atomicAdd(&din[dst[e]], 1u);
    }
}

__global__ void norm_kernel(const unsigned* __restrict__ dout, const unsigned* __restrict__ din,
                            float* __restrict__ ns, float* __restrict__ nd, int N) {
    int i = blockIdx.x * blockDim.x + threadIdx.x;
    if (i < N) {
        unsigned o = dout[i]; if (o < 1u) o = 1u;
        unsigned d = din[i];  if (d < 1u) d = 1u;
        ns[i] = 1.0f / sqrtf((float)o);
        nd[i] = 1.0f / sqrtf((float)d);
    }
}

// ---------------- exclusive scan of in-degrees -> row_ptr ----------------
__global__ __launch_bounds__(1024) void scan1_kernel(const unsigned* __restrict__ deg,
                                                     unsigned* __restrict__ part,
                                                     unsigned* __restrict__ bsum, int N) {
    __shared__ unsigned s[1024];
    int t = threadIdx.x;
    int i = blockIdx.x * 1024 + t;
    unsigned v = (i < N) ? deg[i] : 0u;
    s[t] = v;
    for (int off = 1; off < 1024; off <<= 1) {
        __syncthreads();
        unsigned x = (t >= off) ? s[t - off] : 0u;
        __syncthreads();
        s[t] += x;
    }
    __syncthreads();
    if (i <= N) part[i] = s[t] - v;     // exclusive prefix within block
    if (t == 1023) bsum[blockIdx.x] = s[1023];
}

__global__ void scan2_kernel(unsigned* __restrict__ bsum, int nb) {
    if (threadIdx.x == 0 && blockIdx.x == 0) {
        unsigned run = 0;
        for (int p = 0; p < nb; ++p) { unsigned v = bsum[p]; bsum[p] = run; run += v; }
    }
}

__global__ __launch_bounds__(1024) void scan3_kernel(unsigned* __restrict__ part,
                                                     const unsigned* __restrict__ bsum, int N) {
    int i = blockIdx.x * 1024 + threadIdx.x;
    if (i <= N) part[i] += bsum[blockIdx.x];
}

__global__ void fill_kernel(const int* __restrict__ src, const int* __restrict__ dst,
                            const unsigned* __restrict__ row_ptr, unsigned* __restrict__ fill,
                            unsigned* __restrict__ eidx, int E) {
    int e = blockIdx.x * blockDim.x + threadIdx.x;
    if (e < E) {
        int d = dst[e];
        unsigned pos = row_ptr[d] + atomicAdd(&fill[d], 1u);
        eidx[pos] = (unsigned)src[e];
    }
}

// ---------------- fp32 WMMA GEMM:  out = (X @ W) * norm_src[:,None] ----------------
// X:[N,Kdim], W:[Kdim,128], out:[N,128].  16-row tile per block, one 16x16 col tile per wave.
__global__ __launch_bounds__(256) void gemm_norm_kernel(const float* __restrict__ X,
                                                        const float* __restrict__ W,
                                                        const float* __restrict__ nrm,
                                                        float* __restrict__ out, int Kdim) {
    const int row0  = blockIdx.x * 16;
    const int wave  = threadIdx.x >> 5;       // 0..7 -> 16-col tile
    const int lane  = threadIdx.x & 31;
    const int col0  = wave * 16;
    const int m     = lane & 15;
    const int khalf = (lane >> 4) * 2;        // 0 or 2

    v8f acc = {};
    for (int k = 0; k < Kdim; k += 4) {
        v2f a, b;
        const float* xr = X + (size_t)(row0 + m) * Kdim + (k + khalf);
        a.x = xr[0];                          // A[m][k+khalf]
        a.y = xr[1];                          // A[m][k+khalf+1]
        const float* wr = W + (size_t)(k + khalf) * HID + col0 + m;
        b.x = wr[0];                          // B[k+khalf][col]
        b.y = wr[HID];                        // B[k+khalf+1][col]
        acc = __builtin_amdgcn_wmma_f32_16x16x4_f32(false, a, false, b,
                                                    (short)0, acc, false, false);
    }
    const int rbase = row0 + ((lane >> 4) << 3);
#pragma unroll
    for (int i = 0; i < 8; ++i) {
        int r = rbase + i;
        out[(size_t)r * HID + col0 + m] = acc[i] * nrm[r];
    }
}

// ---------------- CSR pull aggregation: out[n] = relu(sum_{e in N(n)} tmp[src] * nd[n] + bias) ----
__global__ __launch_bounds__(256) void agg_kernel(const float* __restrict__ tmp,
                                                  const unsigned* __restrict__ row_ptr,
                                                  const unsigned* __restrict__ eidx,
                                                  const float* __restrict__ nd,
                                                  const float* __restrict__ bias,
                                                  float* __restrict__ out, int N) {
    int node = blockIdx.x * 8 + (threadIdx.x >> 5);
    int lane = threadIdx.x & 31;
    if (node >= N) return;
    unsigned beg = row_ptr[node], end = row_ptr[node + 1];
    float4 acc = make_float4(0.f, 0.f, 0.f, 0.f);
    for (unsigned j = beg; j < end; ++j) {
        unsigned s = eidx[j];
        const float4 v = *(const float4*)(tmp + (size_t)s * HID + lane * 4);
        acc.x += v.x; acc.y += v.y; acc.z += v.z; acc.w += v.w;
    }
    float sc = nd[node];
    const float4 b = *(const float4*)(bias + lane * 4);
    float4 r;
    r.x = fmaxf(acc.x * sc + b.x, 0.f);
    r.y = fmaxf(acc.y * sc + b.y, 0.f);
    r.z = fmaxf(acc.z * sc + b.z, 0.f);
    r.w = fmaxf(acc.w * sc + b.w, 0.f);
    *(float4*)(out + (size_t)node * HID + lane * 4) = r;
}

// ---------------- sort pool: per-graph bitonic sort (desc key, asc idx), keep top K ----------------
__global__ __launch_bounds__(1024) void topk_kernel(const float* __restrict__ h,
                                                    unsigned* __restrict__ topk, int nper) {
    __shared__ unsigned long long keys[SORTN];   // 128 KB
    const int b = blockIdx.x;
    const int t = threadIdx.x;
    for (int i = t; i < SORTN; i += 1024) {
        unsigned long long v;
        if (i < nper) {
            float kf = h[((size_t)(b * nper + i)) * HID + (HID - 1)];
            unsigned u   = __float_as_uint(kf);
            unsigned ord = (u & 0x80000000u) ? ~u : (u | 0x80000000u); // monotone asc map
            unsigned inv = ~ord;                                        // descending
            v = ((unsigned long long)inv << 32) | (unsigned)i;
        } else {
            v = ~0ull;                                                  // pad -> sorts last
        }
        keys[i] = v;
    }
    __syncthreads();
    for (int size = 2; size <= SORTN; size <<= 1) {
        for (int stride = size >> 1; stride > 0; stride >>= 1) {
            for (int p = t; p < SORTN / 2; p += 1024) {
                int i = 2 * p - (p & (stride - 1));
                int j = i + stride;
                bool asc = ((i & size) == 0);
                unsigned long long a = keys[i], c = keys[j];
                if ((a > c) == asc) { keys[i] = c; keys[j] = a; }
            }
            __syncthreads();
        }
    }
    for (int r = t; r < KSEL; r += 1024)
        topk[b * KSEL + r] = (unsigned)(keys[r] & 0xFFFFFFFFu);
}

// ---------------- classifier ----------------
__global__ void init_out_kernel(const float* __restrict__ b3, float* __restrict__ out) {
    int i = blockIdx.x * blockDim.x + threadIdx.x;
    if (i < 8 * NCLS) out[i] = b3[i % NCLS];
}

__global__ __launch_bounds__(256) void cls_kernel(const float* __restrict__ h,
                                                  const unsigned* __restrict__ topk,
                                                  const float* __restrict__ W3,
                                                  float* __restrict__ out, int nper) {
    const int b     = blockIdx.y;
    const int kbase = blockIdx.x * 125;          // 40 chunks * 125 = KSEL
    const int t     = threadIdx.x;
    __shared__ float lacc[NCLS];
    if (t < NCLS) lacc[t] = 0.f;
    __syncthreads();

    float acc[NCLS];
#pragma unroll
    for (int c = 0; c < NCLS; ++c) acc[c] = 0.f;

    for (int e = t; e < 125 * HID; e += 256) {
        int k  = kbase + (e >> 7);
        int ch = e & (HID - 1);
        unsigned li = topk[b * KSEL + k];
        float x = h[((size_t)(b * nper + (int)li)) * HID + ch];
        const float* wr = W3 + ((size_t)(k * HID + ch)) * NCLS;
#pragma unroll
        for (int c = 0; c < NCLS; ++c) acc[c] += x * wr[c];
    }
#pragma unroll
    for (int c = 0; c < NCLS; ++c) atomicAdd(&lacc[c], acc[c]);
    __syncthreads();
    if (t < NCLS) atomicAdd(&out[b * NCLS + t], lacc[t]);
}

// ---------------- host ----------------
extern "C" void kernel_launch(void* const* d_in, const int* in_sizes, int n_in,
                              void* d_out, int out_size, void* d_ws, size_t ws_size,
                              hipStream_t stream) {
    const float* features = (const float*)d_in[0];
    const int*   src      = (const int*)  d_in[1];
    const int*   dst      = (const int*)  d_in[2];
    const float* W1       = (const float*)d_in[3];
    const float* b1       = (const float*)d_in[4];
    const float* W2       = (const float*)d_in[5];
    const float* b2       = (const float*)d_in[6];
    const float* W3       = (const float*)d_in[7];
    const float* b3       = (const float*)d_in[8];
    float* out = (float*)d_out;

    const int N    = in_sizes[0] / 64;    // IN_DIM = 64
    const int E    = in_sizes[1];
    const int nper = N / 8;               // nodes per graph

    // carve 256B-aligned sections from d_ws
    char* base = (char*)d_ws;
    size_t cur = 0;
    auto take = [&](size_t nbytes) -> char* {
        char* p = base + cur;
        cur += (nbytes + 255) & ~(size_t)255;
        return p;
    };
    unsigned* deg_out = (unsigned*)take((size_t)N * 4);
    unsigned* deg_in  = (unsigned*)take((size_t)N * 4);
    unsigned* fill    = (unsigned*)take((size_t)N * 4);
    unsigned* row_ptr = (unsigned*)take((size_t)(N + 1) * 4);
    unsigned* bsum    = (unsigned*)take(1024 * 4);
    float*    nsrc    = (float*)   take((size_t)N * 4);
    float*    ndst    = (float*)   take((size_t)N * 4);
    unsigned* topk    = (unsigned*)take((size_t)8 * KSEL * 4);
    unsigned* eidx    = (unsigned*)take((size_t)E * 4);
    float*    tmp     = (float*)   take((size_t)N * HID * 4);
    float*    hbuf    = (float*)   take((size_t)N * HID * 4);
    (void)ws_size; (void)n_in; (void)out_size;

    const int tpb = 256;
    // CSR build (once; reused by both conv layers)
    zero_u32_kernel<<<(N + tpb - 1) / tpb, tpb, 0, stream>>>(deg_out, N);
    zero_u32_kernel<<<(N + tpb - 1) / tpb, tpb, 0, stream>>>(deg_in, N);
    zero_u32_kernel<<<(N + tpb - 1) / tpb, tpb, 0, stream>>>(fill, N);
    deg_kernel<<<(E + tpb - 1) / tpb, tpb, 0, stream>>>(src, dst, deg_out, deg_in, E);
    norm_kernel<<<(N + tpb - 1) / tpb, tpb, 0, stream>>>(deg_out, deg_in, nsrc, ndst, N);

    const int nb = (N + 1024) / 1024;     // nb*1024 >= N+1
    scan1_kernel<<<nb, 1024, 0, stream>>>(deg_in, row_ptr, bsum, N);
    scan2_kernel<<<1, 32, 0, stream>>>(bsum, nb);
    scan3_kernel<<<nb, 1024, 0, stream>>>(row_ptr, bsum, N);
    fill_kernel<<<(E + tpb - 1) / tpb, tpb, 0, stream>>>(src, dst, row_ptr, fill, eidx, E);

    // layer 1: tmp = (X @ W1) * nsrc ; hbuf = relu(agg * ndst + b1)
    gemm_norm_kernel<<<N / 16, 256, 0, stream>>>(features, W1, nsrc, tmp, 64);
    agg_kernel<<<(N + 7) / 8, 256, 0, stream>>>(tmp, row_ptr, eidx, ndst, b1, hbuf, N);
    // layer 2
    gemm_norm_kernel<<<N / 16, 256, 0, stream>>>(hbuf, W2, nsrc, tmp, 128);
    agg_kernel<<<(N + 7) / 8, 256, 0, stream>>>(tmp, row_ptr, eidx, ndst, b2, hbuf, N);

    // sort pool + classifier
    topk_kernel<<<8, 1024, 0, stream>>>(hbuf, topk, nper);
    init_out_kernel<<<1, 256, 0, stream>>>(b3, out);
    dim3 cgrid(KSEL / 125, 8);
    cls_kernel<<<cgrid, 256, 0, stream>>>(hbuf, topk, W3, out, nper);
}